// GaussianSplatting_47562467836190
// MI455X (gfx1250) — compile-verified
//
#include <hip/hip_runtime.h>
#include <hip/hip_bf16.h>
#include <stdint.h>

// -----------------------------------------------------------------------------
// GaussianSplatting: out = concat(means[3N], param_colors[3N], cov[9N], opac[N])
// Memory-bound streaming kernel: 224 MB read + 256 MB write -> ~20.6 us floor
// at 23.3 TB/s. cov = (R(q)*diag(s)) (R(q)*diag(s))^T per point (~80 FLOPs,
// i.e. 0.32 GFLOP total -> compute is free; WMMA would only add lane-shuffle
// overhead for 3x3 block-diagonal products). The chip-rewarded path is pure
// data movement: b128 transactions everywhere, strided inputs/outputs staged
// through LDS with CDNA5 async global<->LDS transfers, and non-temporal cache
// hints since the 480 MB one-pass stream is 2.5x the 192 MB L2.
// -----------------------------------------------------------------------------

#define TPB 256          // 8 wave32 per block
#define WAVES_PER_BLOCK (TPB / 32)

typedef float v4f __attribute__((ext_vector_type(4)));

__global__ __launch_bounds__(TPB)
void gs_cov_kernel(const v4f*   __restrict__ quats,      // (N,4) [w,x,y,z]
                   const float* __restrict__ scales,     // (N,3)
                   float*       __restrict__ cov_out,    // (N,9)
                   int N)
{
    // Per-wave staging buffers (16B aligned for b128 async transfers).
    __shared__ __align__(16) float lds_c[WAVES_PER_BLOCK][288];  // 32 pts x 9 cov
    __shared__ __align__(16) float lds_s[WAVES_PER_BLOCK][96];   // 32 pts x 3 scales

    const int tid   = threadIdx.x;
    const int wave  = tid >> 5;
    const int lane  = tid & 31;
    const int gid   = blockIdx.x * TPB + tid;   // point index
    const int wbase = gid - lane;               // first point of this wave
    const bool full = (wbase + 32 <= N);        // wave-uniform predicate

    float s0, s1, s2;
    v4f q = (v4f){1.f, 0.f, 0.f, 0.f};

    if (full) {
        // -- Stage this wave's 96 scale floats: 24 x b128 async loads to LDS --
        if (lane < 24) {
            uint32_t loff = (uint32_t)(uintptr_t)(&lds_s[wave][lane * 4]);
            uint64_t ga   = (uint64_t)(uintptr_t)(scales + (size_t)wbase * 3 + (size_t)lane * 4);
            asm volatile("global_load_async_to_lds_b128 %0, %1, off th:TH_LOAD_NT"
                         :: "v"(loff), "v"(ga) : "memory");
        }
        // -- Coalesced 16B quat load overlaps the async LDS fill (no guard:
        //    full wave => all lanes valid, no divergence) ---------------------
        q = __builtin_nontemporal_load(quats + gid);

        asm volatile("s_wait_asynccnt 0" ::: "memory");
        s0 = lds_s[wave][lane * 3 + 0];   // stride-3 dwords: gcd(3,64)=1 -> conflict-free
        s1 = lds_s[wave][lane * 3 + 1];
        s2 = lds_s[wave][lane * 3 + 2];
    } else if (gid < N) {                 // tail wave (never taken when 256|N)
        q = quats[gid];
        const float* sp = scales + (size_t)gid * 3;
        s0 = sp[0]; s1 = sp[1]; s2 = sp[2];
    } else {
        s0 = s1 = s2 = 0.f;
    }

    // ---- Quaternion -> rotation -> covariance (VALU; hidden under HBM) -----
    float a = q.x, b = q.y, c = q.z, d = q.w;       // w, x, y, z
    float inv = rsqrtf(a * a + b * b + c * c + d * d);
    a *= inv; b *= inv; c *= inv; d *= inv;

    float r00 = 1.f - 2.f * (c * c + d * d);
    float r01 = 2.f * (b * c - a * d);
    float r02 = 2.f * (b * d + a * c);
    float r10 = 2.f * (b * c + a * d);
    float r11 = 1.f - 2.f * (b * b + d * d);
    float r12 = 2.f * (c * d - a * b);
    float r20 = 2.f * (b * d - a * c);
    float r21 = 2.f * (c * d + a * b);
    float r22 = 1.f - 2.f * (b * b + c * c);

    float t0 = s0 * s0, t1 = s1 * s1, t2 = s2 * s2;  // cov = R diag(s^2) R^T
    float c00 = r00 * r00 * t0 + r01 * r01 * t1 + r02 * r02 * t2;
    float c01 = r00 * r10 * t0 + r01 * r11 * t1 + r02 * r12 * t2;
    float c02 = r00 * r20 * t0 + r01 * r21 * t1 + r02 * r22 * t2;
    float c11 = r10 * r10 * t0 + r11 * r11 * t1 + r12 * r12 * t2;
    float c12 = r10 * r20 * t0 + r11 * r21 * t1 + r12 * r22 * t2;
    float c22 = r20 * r20 * t0 + r21 * r21 * t1 + r22 * r22 * t2;

    if (full) {
        // stride-9 dwords across lanes: gcd(9,64)=1 -> conflict-free banks
        float* wp = &lds_c[wave][lane * 9];
        wp[0] = c00; wp[1] = c01; wp[2] = c02;
        wp[3] = c01; wp[4] = c11; wp[5] = c12;
        wp[6] = c02; wp[7] = c12; wp[8] = c22;

        // Our wave's LDS writes must land before the async engine reads them.
        asm volatile("s_wait_dscnt 0" ::: "memory");

        // 288 contiguous floats -> 72 x b128 async NT stores (fully coalesced
        // 1152 B burst per wave instead of 9 stride-36B scalar stores/lane).
        const size_t obase = (size_t)wbase * 9;
        {
            uint32_t loff = (uint32_t)(uintptr_t)(&lds_c[wave][lane * 4]);
            uint64_t ga   = (uint64_t)(uintptr_t)(cov_out + obase + (size_t)lane * 4);
            asm volatile("global_store_async_from_lds_b128 %0, %1, off th:TH_STORE_NT"
                         :: "v"(ga), "v"(loff) : "memory");
        }
        {
            const int ch  = lane + 32;
            uint32_t loff = (uint32_t)(uintptr_t)(&lds_c[wave][ch * 4]);
            uint64_t ga   = (uint64_t)(uintptr_t)(cov_out + obase + (size_t)ch * 4);
            asm volatile("global_store_async_from_lds_b128 %0, %1, off th:TH_STORE_NT"
                         :: "v"(ga), "v"(loff) : "memory");
        }
        if (lane < 8) {
            const int ch  = lane + 64;
            uint32_t loff = (uint32_t)(uintptr_t)(&lds_c[wave][ch * 4]);
            uint64_t ga   = (uint64_t)(uintptr_t)(cov_out + obase + (size_t)ch * 4);
            asm volatile("global_store_async_from_lds_b128 %0, %1, off th:TH_STORE_NT"
                         :: "v"(ga), "v"(loff) : "memory");
        }
        // Drain before LDS is released (s_endpgm also waits; explicit is safe).
        asm volatile("s_wait_asynccnt 0" ::: "memory");
    } else if (gid < N) {
        float* o = cov_out + (size_t)gid * 9;
        o[0] = c00; o[1] = c01; o[2] = c02;
        o[3] = c01; o[4] = c11; o[5] = c12;
        o[6] = c02; o[7] = c12; o[8] = c22;
    }
}

extern "C" void kernel_launch(void* const* d_in, const int* in_sizes, int n_in,
                              void* d_out, int out_size, void* d_ws, size_t ws_size,
                              hipStream_t stream) {
    // setup_inputs order:
    // 0: points, 1: colors, 2: scales, 3: means, 4: param_scales,
    // 5: quats, 6: param_colors, 7: opacities
    const float* means        = (const float*)d_in[3];
    const float* param_scales = (const float*)d_in[4];
    const v4f*   quats        = (const v4f*)  d_in[5];
    const float* param_colors = (const float*)d_in[6];
    const float* opacities    = (const float*)d_in[7];
    const int N = in_sizes[7];
    if (N <= 0) return;

    float* out = (float*)d_out;
    const size_t N3 = (size_t)N * 3;

    // Outputs 0,1,3 are verbatim copies of inputs: full-bandwidth D2D copies
    // (SDMA, overlapping the compute kernel's HBM stream under graph replay).
    hipMemcpyAsync(out,                  means,        N3 * sizeof(float),
                   hipMemcpyDeviceToDevice, stream);
    hipMemcpyAsync(out + N3,             param_colors, N3 * sizeof(float),
                   hipMemcpyDeviceToDevice, stream);
    hipMemcpyAsync(out + (size_t)15 * N, opacities,    (size_t)N * sizeof(float),
                   hipMemcpyDeviceToDevice, stream);

    // Output 2: covariance, computed + streamed via async LDS<->global b128.
    float* cov_out = out + (size_t)6 * N;
    const int blocks = (N + TPB - 1) / TPB;   // N=4M -> 15625 full blocks
    gs_cov_kernel<<<blocks, TPB, 0, stream>>>(quats, param_scales, cov_out, N);
}